// AgentLoss_70866960384004
// MI455X (gfx1250) — compile-verified
//
#include <hip/hip_runtime.h>
#include <hip/hip_bf16.h>

#define GAMMA        0.99f
#define GAE_LAMBDA   0.95f
#define ENTROPY_COEF 0.01f
#define VALUE_COEF   0.5f

typedef __attribute__((ext_vector_type(2))) float v2f;
typedef __attribute__((ext_vector_type(4))) float v4f;
typedef __attribute__((ext_vector_type(8))) float v8f;
typedef __attribute__((ext_vector_type(2))) int   v2i;

static constexpr int       T_LEN = 65536;
static constexpr long long S_N   = 1048576;
static constexpr int       A_N   = 6;

// Fixed launch geometry -> compile-time uniform trip counts (full EXEC at WMMA).
static constexpr int GRID_X  = 256;
static constexpr int BLOCK   = 256;
static constexpr long long STRIDE = (long long)GRID_X * BLOCK;   // 65536
static constexpr int TRIPS   = (int)((S_N / 2) / STRIDE);        // 8

// ---------------------------------------------------------------------------
// WMMA accumulate: A = ones(16x4), B = 2 contributions/lane (64/wave).
// D[m][n] = colsum(B)[n] + C[m][n]; each value counted 16x -> /16 at commit.
// Called only from straight-line code with uniform control flow (EXEC all-1s).
// ---------------------------------------------------------------------------
__device__ __forceinline__ v8f wmma_acc(v8f acc, float c0, float c1) {
#if __has_builtin(__builtin_amdgcn_wmma_f32_16x16x4_f32)
    v2f a; a[0] = 1.0f; a[1] = 1.0f;
    v2f b; b[0] = c0;   b[1] = c1;
    return __builtin_amdgcn_wmma_f32_16x16x4_f32(
        false, a, false, b, (short)0, acc, false, false);
#else
    acc[0] += c0 + c1;
    return acc;
#endif
}

__device__ __forceinline__ void commit_partial(v8f acc, float* gdst) {
    float p = acc[0] + acc[1] + acc[2] + acc[3] +
              acc[4] + acc[5] + acc[6] + acc[7];
    for (int off = 16; off > 0; off >>= 1)          // wave32 reduce
        p += __shfl_down(p, off, 32);
    if ((threadIdx.x & 31) == 0)
        atomicAdd(gdst, p * (1.0f / 16.0f));
}

// ---------------------------------------------------------------------------
// CE from 6 register-resident log-probs; target gathered via cndmask selects.
// ---------------------------------------------------------------------------
__device__ __forceinline__ float ce_regs(float f0, float f1, float f2,
                                         float f3, float f4, float f5, int act) {
    float m = fmaxf(fmaxf(fmaxf(f0, f1), fmaxf(f2, f3)), fmaxf(f4, f5));
    float s = __expf(f0 - m) + __expf(f1 - m) + __expf(f2 - m) +
              __expf(f3 - m) + __expf(f4 - m) + __expf(f5 - m);
    float logZ = m + __logf(s);
    float t = f0;
    t = (act == 1) ? f1 : t;
    t = (act == 2) ? f2 : t;
    t = (act == 3) ? f3 : t;
    t = (act == 4) ? f4 : t;
    t = (act == 5) ? f5 : t;
    return logZ - t;
}

// ---------------------------------------------------------------------------
// Opponent cross-entropy over (O, S, 6): 192 MB single-pass stream.
// Pair of elements = 12 floats = 3 x b128 non-temporal loads (16B aligned).
// ---------------------------------------------------------------------------
__global__ void opp_ce_kernel(const float* __restrict__ logp,
                              const int*   __restrict__ acts,
                              const float* __restrict__ coefs,
                              float*       __restrict__ ws) {
    const int o = blockIdx.y;
    const float scale = coefs[o] * (1.0f / (float)S_N);
    const v4f* __restrict__ qbase =
        reinterpret_cast<const v4f*>(logp + (size_t)o * S_N * A_N);
    const v2i* __restrict__ abase =
        reinterpret_cast<const v2i*>(acts + (size_t)o * S_N);

    long long i = (long long)blockIdx.x * BLOCK + threadIdx.x;
    v8f acc = {};
    for (int it = 0; it < TRIPS; ++it, i += STRIDE) {   // uniform trip count
        const v4f* q = qbase + 3 * i;
        __builtin_prefetch(q + 3 * STRIDE, 0, 1);
        v4f q0 = __builtin_nontemporal_load(q);
        v4f q1 = __builtin_nontemporal_load(q + 1);
        v4f q2 = __builtin_nontemporal_load(q + 2);
        v2i aa = __builtin_nontemporal_load(abase + i);
        float c0 = scale * ce_regs(q0.x, q0.y, q0.z, q0.w, q1.x, q1.y, aa.x);
        float c1 = scale * ce_regs(q1.z, q1.w, q2.x, q2.y, q2.z, q2.w, aa.y);
        acc = wmma_acc(acc, c0, c1);                    // full-EXEC site
    }
    commit_partial(acc, ws);
}

// ---------------------------------------------------------------------------
// Opponent smooth-L1 value loss over (O, S-1).
// ---------------------------------------------------------------------------
__global__ void opp_val_kernel(const float* __restrict__ vals,
                               const float* __restrict__ rews,
                               const float* __restrict__ coefs,
                               float*       __restrict__ ws) {
    const int o = blockIdx.y;
    const float scale = coefs[o] * (1.0f / (float)(S_N - 1));
    const float* __restrict__ vb = vals + (size_t)o * S_N;
    const float* __restrict__ rb = rews + (size_t)o * (S_N - 1);

    long long i = (long long)blockIdx.x * BLOCK + threadIdx.x;
    v8f acc = {};
    for (int it = 0; it < TRIPS; ++it, i += STRIDE) {   // uniform trip count
        const long long s0 = 2 * i;
        // v2 = {vals[s0], vals[s0+1]} : 8B aligned, always in range.
        v2f v2 = __builtin_nontemporal_load(
            reinterpret_cast<const v2f*>(vb + s0));
        float r0 = __builtin_nontemporal_load(rb + s0);
        float d0  = v2.x - (r0 + GAMMA * v2.y);
        float ad0 = fabsf(d0);
        float c0  = scale * (ad0 < 1.0f ? 0.5f * d0 * d0 : ad0 - 0.5f);
        float c1  = 0.0f;
        if (s0 + 1 < S_N - 1) {                         // tail guard (data only)
            float vnn = __builtin_nontemporal_load(vb + s0 + 2);
            float r1  = __builtin_nontemporal_load(rb + s0 + 1);
            float d1  = v2.y - (r1 + GAMMA * vnn);
            float ad1 = fabsf(d1);
            c1 = scale * (ad1 < 1.0f ? 0.5f * d1 * d1 : ad1 - 0.5f);
        }
        acc = wmma_acc(acc, c0, c1);                    // full-EXEC site
    }
    commit_partial(acc, ws);
}

// ---------------------------------------------------------------------------
// Agent loss: blocked parallel scan of two linear reverse recurrences.
// 256 chunks x 256 elems; summaries, serial combine, exact replay.
// ---------------------------------------------------------------------------
__global__ void agent_kernel(const float* __restrict__ r,
                             const float* __restrict__ lp,
                             const float* __restrict__ v,
                             const float* __restrict__ ent,
                             float*       __restrict__ ws) {
    constexpr int NB = 256, CH = 256;
    __shared__ float Br[NB], Bg[NB], bndR[NB], bndG[NB], red[NB];

    const int t = threadIdx.x;
    const int s = t * CH, e = s + CH;

    float br = 0.0f, bg = 0.0f;
    for (int i = e - 1; i >= s; --i) {
        br = GAMMA * br + r[i];
        float delta = r[i] + GAMMA * v[i + 1] - v[i];
        bg = (GAMMA * GAE_LAMBDA) * bg + delta;
    }
    Br[t] = br; Bg[t] = bg;
    __syncthreads();

    if (t == 0) {
        const float Ar = __powf(GAMMA, (float)CH);
        const float Ag = __powf(GAMMA * GAE_LAMBDA, (float)CH);
        float R = v[T_LEN], G = 0.0f;
        for (int c = NB - 1; c >= 0; --c) {
            bndR[c] = R; bndG[c] = G;
            R = Ar * R + Br[c];
            G = Ag * G + Bg[c];
        }
    }
    __syncthreads();

    float R = bndR[t], G = bndG[t], pl = 0.0f, vl = 0.0f;
    for (int i = e - 1; i >= s; --i) {
        const float ri = r[i], vi = v[i];
        R = GAMMA * R + ri;
        const float adv = R - vi;
        vl += 0.5f * adv * adv;
        const float delta = ri + GAMMA * v[i + 1] - vi;
        G = G * (GAMMA * GAE_LAMBDA) + delta;
        pl += -lp[i] * G - ENTROPY_COEF * ent[i];
    }
    red[t] = pl + VALUE_COEF * vl;
    __syncthreads();
    for (int off = NB / 2; off > 0; off >>= 1) {
        if (t < off) red[t] += red[t + off];
        __syncthreads();
    }
    if (t == 0) ws[1] = red[0];
}

__global__ void finalize_kernel(const float* __restrict__ ws,
                                float* __restrict__ out) {
    out[0] = ws[0] + ws[1];
}

// ---------------------------------------------------------------------------
extern "C" void kernel_launch(void* const* d_in, const int* in_sizes, int n_in,
                              void* d_out, int out_size, void* d_ws, size_t ws_size,
                              hipStream_t stream) {
    (void)in_sizes; (void)n_in; (void)out_size; (void)ws_size;
    const float* a_r   = (const float*)d_in[0];   // agent_rewards (T)
    const float* a_lp  = (const float*)d_in[1];   // agent_log_probs (T)
    const float* a_v   = (const float*)d_in[2];   // agent_values (T+1)
    const float* a_ent = (const float*)d_in[3];   // agent_entropies (T)
    const float* o_lp  = (const float*)d_in[4];   // opponent_log_probs (O,S,A)
    const int*   o_act = (const int*)  d_in[5];   // opponent_actions (O,S)
    const float* o_v   = (const float*)d_in[6];   // opponent_values (O,S)
    const float* o_r   = (const float*)d_in[7];   // opponent_rewards (O,S-1)
    const float* o_c   = (const float*)d_in[8];   // opponent_coefs (O)
    float* ws  = (float*)d_ws;
    float* out = (float*)d_out;

    hipMemsetAsync(ws, 0, 2 * sizeof(float), stream);  // ws[0]=opp, ws[1]=agent

    dim3 grid(GRID_X, 8, 1);
    opp_ce_kernel <<<grid, BLOCK, 0, stream>>>(o_lp, o_act, o_c, ws);
    opp_val_kernel<<<grid, BLOCK, 0, stream>>>(o_v,  o_r,  o_c, ws);
    agent_kernel  <<<1,    BLOCK, 0, stream>>>(a_r, a_lp, a_v, a_ent, ws);
    finalize_kernel<<<1, 1, 0, stream>>>(ws, out);
}